// OptimizedInvariantMACE_62045097558035
// MI455X (gfx1250) — compile-verified
//
#include <hip/hip_runtime.h>
#include <cstddef>
#include <cstdint>

typedef __attribute__((ext_vector_type(2))) float v2f;
typedef __attribute__((ext_vector_type(8))) float v8f;

#define C_CH 128
#define LM   16

__device__ __forceinline__ int l_of_lm(int lm) {
    return (lm == 0) ? 0 : (lm < 4) ? 1 : (lm < 9) ? 2 : 3;
}

// CDNA5 async global->LDS copy (bypasses VGPRs, tracked by ASYNCcnt)
__device__ __forceinline__ void async_copy_b128(unsigned lds_off, const float* gaddr) {
    asm volatile("global_load_async_to_lds_b128 %0, %1, off"
                 :: "v"(lds_off), "v"((unsigned long long)(uintptr_t)gaddr)
                 : "memory");
}
__device__ __forceinline__ void wait_asynccnt0() {
    asm volatile("s_wait_asynccnt 0x0" ::: "memory");
}

// ---------------------------------------------------------------------------
// 0) zero the msg accumulator region of the workspace
// ---------------------------------------------------------------------------
__global__ __launch_bounds__(256) void zero_kernel(float* __restrict__ p, size_t n) {
    size_t i = (size_t)blockIdx.x * blockDim.x + threadIdx.x;
    size_t stride = (size_t)gridDim.x * blockDim.x;
    for (; i < n; i += stride) p[i] = 0.0f;
}

// ---------------------------------------------------------------------------
// 1) h = node_feats @ W_up    [N,128] @ [128,128], f32 WMMA 16x16x4
//    block = 256 threads = 8 waves; wave w owns col-tile w; block owns 16 rows
// ---------------------------------------------------------------------------
__global__ __launch_bounds__(256) void linear_up_kernel(const float* __restrict__ NF,
                                                        const float* __restrict__ W,
                                                        float* __restrict__ H, int N) {
    const int wave = threadIdx.x >> 5;
    const int lane = threadIdx.x & 31;
    const int row0 = blockIdx.x * 16;
    const int col0 = wave * 16;
    const int m  = lane & 15;        // A-row / B-col within tile
    const int kh = (lane >> 4) * 2;  // K-half select (0 or 2)

    int arow = row0 + m;
    if (arow >= N) arow = N - 1;     // clamp loads; stores are guarded

    v8f acc = {};
    for (int k = 0; k < C_CH; k += 4) {
        v2f a, b;
        a.x = NF[(size_t)arow * C_CH + k + kh];
        a.y = NF[(size_t)arow * C_CH + k + kh + 1];
        b.x = W[(size_t)(k + kh)     * C_CH + col0 + m];
        b.y = W[(size_t)(k + kh + 1) * C_CH + col0 + m];
        acc = __builtin_amdgcn_wmma_f32_16x16x4_f32(false, a, false, b,
                                                    (short)0, acc, false, false);
    }
    const int rowOff = (lane >> 4) * 8;
#pragma unroll
    for (int r = 0; r < 8; ++r) {
        int row = row0 + rowOff + r;
        if (row < N) H[(size_t)row * C_CH + col0 + m] = acc[r];
    }
}

// ---------------------------------------------------------------------------
// 2) W_comb[e,l] = (W_lin[l] @ W_skip[e,l]) / AVG_NUM_NEIGHBORS
//    grid = NE*4 blocks; block = 8 waves; wave w = col-tile, loops 8 row-tiles
// ---------------------------------------------------------------------------
__global__ __launch_bounds__(256) void combine_weights_kernel(const float* __restrict__ WLIN,
                                                              const float* __restrict__ WSKIP,
                                                              float* __restrict__ WC) {
    const int e = blockIdx.x >> 2;
    const int l = blockIdx.x & 3;
    const float* WL = WLIN  + (size_t)l * C_CH * C_CH;
    const float* WS = WSKIP + ((size_t)e * 4 + l) * C_CH * C_CH;
    float*       WO = WC    + ((size_t)e * 4 + l) * C_CH * C_CH;

    const int wave = threadIdx.x >> 5;
    const int lane = threadIdx.x & 31;
    const int col0 = wave * 16;
    const int m  = lane & 15;
    const int kh = (lane >> 4) * 2;
    const int rowOff = (lane >> 4) * 8;
    const float scale = 1.0f / 16.0f;   // AVG_NUM_NEIGHBORS

    for (int rt = 0; rt < 8; ++rt) {
        const int row0 = rt * 16;
        v8f acc = {};
        for (int k = 0; k < C_CH; k += 4) {
            v2f a, b;
            a.x = WL[(size_t)(row0 + m) * C_CH + k + kh];
            a.y = WL[(size_t)(row0 + m) * C_CH + k + kh + 1];
            b.x = WS[(size_t)(k + kh)     * C_CH + col0 + m];
            b.y = WS[(size_t)(k + kh + 1) * C_CH + col0 + m];
            acc = __builtin_amdgcn_wmma_f32_16x16x4_f32(false, a, false, b,
                                                        (short)0, acc, false, false);
        }
#pragma unroll
        for (int r = 0; r < 8; ++r)
            WO[(size_t)(row0 + rowOff + r) * C_CH + col0 + m] = acc[r] * scale;
    }
}

// ---------------------------------------------------------------------------
// 3) edge message + scatter:  msg[recv] += h[send]*attr[lm]*ef[l(lm)]
//    one block per edge, 256 threads; atomics resolve in L2 (msg = 41MB << 192MB)
// ---------------------------------------------------------------------------
__global__ __launch_bounds__(256) void edge_message_kernel(const float* __restrict__ H,
                                                           const float* __restrict__ ATTR,
                                                           const float* __restrict__ EF,
                                                           const int*   __restrict__ EIDX,
                                                           float* __restrict__ MSG,
                                                           int E) {
    const int e = blockIdx.x;
    if (e >= E) return;
    const int s = EIDX[e];        // sender   = edge_index[0][e]
    const int r = EIDX[E + e];    // receiver = edge_index[1][e]

    const int t   = threadIdx.x;
    const int c   = t & 127;
    const int lm0 = (t >> 7) * 8; // 0 or 8

    const float* efE = EF   + (size_t)e * 4 * C_CH;
    const float* atE = ATTR + (size_t)e * LM;
    __builtin_prefetch(&efE[c], 0, 3);          // global_prefetch_b8: pull ef row

    const float hv = H[(size_t)s * C_CH + c];
    float*       mr  = MSG  + (size_t)r * LM * C_CH;

#pragma unroll
    for (int j = 0; j < 8; ++j) {
        const int lm = lm0 + j;
        const int l  = l_of_lm(lm);
        const float v = hv * atE[lm] * efE[(size_t)l * C_CH + c];
        unsafeAtomicAdd(&mr[(size_t)lm * C_CH + c], v);   // global_atomic_add_f32
    }
}

// ---------------------------------------------------------------------------
// 4) per-node transform: out[n] = msg[n] @ W_comb[elem[n], l(row)]
//    block per node; msg[n] (16x128 = 8KB) staged into LDS via CDNA5 async
//    global->LDS copies (no VGPR roundtrip); 4 l-masked WMMA passes
// ---------------------------------------------------------------------------
__global__ __launch_bounds__(256) void node_transform_kernel(const float* __restrict__ MSG,
                                                             const float* __restrict__ WC,
                                                             const int*   __restrict__ ELEM,
                                                             float* __restrict__ OUT,
                                                             int N) {
    __shared__ float sm[LM * C_CH];   // 8 KB
    const int n = blockIdx.x;
    if (n >= N) return;

    // async-stage msg[n]: 256 threads x 32B = 8KB, two b128 copies per thread
    {
        const float* src = MSG + (size_t)n * LM * C_CH + (size_t)threadIdx.x * 8;
        unsigned lds_off = (unsigned)(unsigned long long)(uintptr_t)sm
                         + (unsigned)threadIdx.x * 32u;
        async_copy_b128(lds_off,       src);
        async_copy_b128(lds_off + 16u, src + 4);
    }

    const int e = ELEM[n];                       // overlaps with async fill
    const float* W = WC + (size_t)e * 4 * C_CH * C_CH;

    const int wave = threadIdx.x >> 5;
    const int lane = threadIdx.x & 31;
    const int col0 = wave * 16;
    const int m  = lane & 15;          // A-row = lm index
    const int kh = (lane >> 4) * 2;
    const int lrow = l_of_lm(m);       // l of this lane's A-row

    wait_asynccnt0();
    __syncthreads();

    v8f acc = {};
    for (int l = 0; l < 4; ++l) {
        const float* Wl = W + (size_t)l * C_CH * C_CH;
        const float msk = (lrow == l) ? 1.0f : 0.0f;   // A_ext block-diagonal mask
        for (int k = 0; k < C_CH; k += 4) {
            v2f a, b;
            a.x = msk * sm[m * C_CH + k + kh];
            a.y = msk * sm[m * C_CH + k + kh + 1];
            b.x = Wl[(size_t)(k + kh)     * C_CH + col0 + m];
            b.y = Wl[(size_t)(k + kh + 1) * C_CH + col0 + m];
            acc = __builtin_amdgcn_wmma_f32_16x16x4_f32(false, a, false, b,
                                                        (short)0, acc, false, false);
        }
    }
    const int rowOff = (lane >> 4) * 8;
#pragma unroll
    for (int r = 0; r < 8; ++r) {
        const int lm = rowOff + r;
        OUT[(size_t)n * LM * C_CH + (size_t)lm * C_CH + col0 + m] = acc[r];
    }
}

// ---------------------------------------------------------------------------
extern "C" void kernel_launch(void* const* d_in, const int* in_sizes, int n_in,
                              void* d_out, int out_size, void* d_ws, size_t ws_size,
                              hipStream_t stream) {
    const float* node_feats = (const float*)d_in[0];
    const float* edge_attrs = (const float*)d_in[1];
    const float* edge_feats = (const float*)d_in[2];
    const float* W_up       = (const float*)d_in[3];
    const float* W_lin      = (const float*)d_in[4];
    const float* W_skip     = (const float*)d_in[5];
    const int*   edge_index = (const int*)  d_in[6];
    const int*   node_elem  = (const int*)  d_in[7];
    float*       out        = (float*)d_out;

    const int C  = C_CH;
    const int N  = in_sizes[0] / C;
    const int E  = in_sizes[1] / LM;
    const int NE = in_sizes[5] / (4 * C * C);

    // workspace layout
    float* h     = (float*)d_ws;                       // N*C
    float* msg   = h   + (size_t)N * C;                // N*16*C
    float* wcomb = msg + (size_t)N * LM * C;           // NE*4*C*C

    // 0) zero msg accumulator
    zero_kernel<<<2048, 256, 0, stream>>>(msg, (size_t)N * LM * C);

    // 1) h = node_feats @ W_up
    linear_up_kernel<<<(N + 15) / 16, 256, 0, stream>>>(node_feats, W_up, h, N);

    // 2) W_comb = (W_lin @ W_skip) / 16
    combine_weights_kernel<<<NE * 4, 256, 0, stream>>>(W_lin, W_skip, wcomb);

    // 3) edge scatter into msg
    edge_message_kernel<<<E, 256, 0, stream>>>(h, edge_attrs, edge_feats,
                                               edge_index, msg, E);

    // 4) per-node combined per-l channel mixing
    node_transform_kernel<<<N, 256, 0, stream>>>(msg, wcomb, node_elem, out, N);
}